// GNN_34875134443755
// MI455X (gfx1250) — compile-verified
//
#include <hip/hip_runtime.h>
#include <hip/hip_bf16.h>

// ---------------------------------------------------------------------------
// GraphSAGE 2-layer forward for MI455X (gfx1250, wave32).
// fp32 throughout (reference is fp32; workload is gather/scatter BW bound at
// ~2.7 GB / 23.3 TB/s ~ 115us, GEMM is only ~6.5 GFLOP) using
// V_WMMA_F32_16X16X4_F32 for the dense lin_l/lin_r projections.
// ---------------------------------------------------------------------------

#define N_NODES 100000
#define N_EDGES 1600000
#define D_IN    64
#define D_HID   128
#define D_OUT   64

typedef __attribute__((ext_vector_type(2))) float v2f;
typedef __attribute__((ext_vector_type(8))) float v8f;

__device__ __forceinline__ v8f wmma_f32_k4(v2f a, v2f b, v8f c) {
    // D = A(16x4,f32) * B(4x16,f32) + C(16x16,f32); 8-arg form:
    // (neg_a, A, neg_b, B, c_mod, C, reuse_a, reuse_b)
    return __builtin_amdgcn_wmma_f32_16x16x4_f32(
        false, a, false, b, (short)0, c, false, false);
}

// ---------------------------------------------------------------------------
// Zero a float4-aligned region of workspace.
// ---------------------------------------------------------------------------
__global__ __launch_bounds__(256) void zero_f4(float4* __restrict__ p, long long n4) {
    long long i = (long long)blockIdx.x * blockDim.x + threadIdx.x;
    if (i < n4) p[i] = make_float4(0.f, 0.f, 0.f, 0.f);
}

// ---------------------------------------------------------------------------
// deg[dst[e]] += 1  (fp32 so the divide stays in float like the reference)
// ---------------------------------------------------------------------------
__global__ __launch_bounds__(256) void degree_kernel(const int* __restrict__ dst,
                                                     float* __restrict__ deg) {
    int e = blockIdx.x * blockDim.x + threadIdx.x;   // grid sized exactly to E
    atomicAdd(deg + dst[e], 1.0f);
}

// ---------------------------------------------------------------------------
// sum[dst[e], :] += feat[src[e], :]
// D/4 lanes per edge; coalesced float4 gather, 4x fp32 global atomics.
// Grid is sized so every thread maps to a valid (edge, chunk) pair.
// ---------------------------------------------------------------------------
template <int D>
__global__ __launch_bounds__(256) void scatter_add_kernel(
        const float* __restrict__ feat,
        const int*   __restrict__ src,
        const int*   __restrict__ dst,
        float*       __restrict__ sum) {
    constexpr int TPE = D / 4;  // threads per edge
    long long t = (long long)blockIdx.x * blockDim.x + threadIdx.x;
    int e  = (int)(t / TPE);
    int d4 = (int)(t % TPE);
    int s  = src[e];
    int d  = dst[e];
    float4 v = reinterpret_cast<const float4*>(feat + (size_t)s * D)[d4];
    float* p = sum + (size_t)d * D + (size_t)d4 * 4;
    atomicAdd(p + 0, v.x);
    atomicAdd(p + 1, v.y);
    atomicAdd(p + 2, v.z);
    atomicAdd(p + 3, v.w);
}

// ---------------------------------------------------------------------------
// out = act( (sum/max(deg,1)) @ Wl + x @ Wr + b )
// One wave computes one 16x16 output tile; K stepped by 4 with
// v_wmma_f32_16x16x4_f32, both matmuls share a single f32 accumulator.
//
// Fragment layouts (per ISA 7.12.2):
//   A 16x4 f32 : lane holds row m=lane&15, k = {kb, kb+1}, kb = 2*(lane>>4)
//   B 4x16 f32 : lane holds col n=lane&15, k = {kb, kb+1}   (mirrored)
//   C/D 16x16  : vgpr j, lane -> (m = j + 8*(lane>>4), n = lane&15)
//
// All grids are exact multiples (N_NODES%16==0), so EXEC is all-1s at every
// WMMA as required.
// ---------------------------------------------------------------------------
template <int DIN, int DOUT, bool RELU>
__global__ __launch_bounds__(256) void sage_gemm_kernel(
        const float* __restrict__ sum,   // [N, DIN] neighbor sums
        const float* __restrict__ deg,   // [N]
        const float* __restrict__ xin,   // [N, DIN] self features
        const float* __restrict__ Wl,    // [DIN, DOUT]
        const float* __restrict__ Wr,    // [DIN, DOUT]
        const float* __restrict__ bias,  // [DOUT]
        float*       __restrict__ out) { // [N, DOUT]
    const int lane = threadIdx.x & 31;
    const int wave = threadIdx.x >> 5;
    const int tile = blockIdx.x * 8 + wave;       // 8 waves / block
    constexpr int TILES_N = DOUT / 16;
    const int tm = tile / TILES_N;
    const int tn = tile % TILES_N;

    const int mr   = lane & 15;
    const int half = lane >> 4;                   // 0 or 1
    const int kb   = half * 2;                    // lane's k offset in k-step
    const int arow = tm * 16 + mr;                // node row this lane loads (A)
    const int ncol = tn * 16 + mr;                // out col this lane loads (B/C)

    const float inv = 1.0f / fmaxf(deg[arow], 1.0f);
    const float* sA = sum + (size_t)arow * DIN;
    const float* xA = xin + (size_t)arow * DIN;

    v8f acc = {};
#pragma unroll
    for (int k0 = 0; k0 < DIN; k0 += 4) {
        const int k = k0 + kb;
        // aggregated-neighbor term: (sum * inv_deg) @ Wl
        v2f a1 = { sA[k] * inv, sA[k + 1] * inv };
        v2f b1 = { Wl[(size_t)k * DOUT + ncol], Wl[(size_t)(k + 1) * DOUT + ncol] };
        acc = wmma_f32_k4(a1, b1, acc);
        // self term: x @ Wr
        v2f a2 = { xA[k], xA[k + 1] };
        v2f b2 = { Wr[(size_t)k * DOUT + ncol], Wr[(size_t)(k + 1) * DOUT + ncol] };
        acc = wmma_f32_k4(a2, b2, acc);
    }

    const float bb = bias[ncol];
#pragma unroll
    for (int j = 0; j < 8; ++j) {
        const int m = tm * 16 + j + 8 * half;
        float v = acc[j] + bb;
        if (RELU) v = fmaxf(v, 0.0f);
        out[(size_t)m * DOUT + ncol] = v;
    }
}

// ---------------------------------------------------------------------------
// Row softmax over 64 logits; one wave per node, 2 elems/lane, shfl reductions.
// ---------------------------------------------------------------------------
__global__ __launch_bounds__(256) void softmax64_kernel(const float* __restrict__ in,
                                                        float* __restrict__ out) {
    const int lane = threadIdx.x & 31;
    const int wave = threadIdx.x >> 5;
    const int node = blockIdx.x * 8 + wave;       // grid sized exactly to N
    const float* row = in + (size_t)node * 64;

    float v0 = row[lane];
    float v1 = row[lane + 32];
    float m = fmaxf(v0, v1);
#pragma unroll
    for (int off = 16; off > 0; off >>= 1)
        m = fmaxf(m, __shfl_xor(m, off, 32));
    float e0 = expf(v0 - m);
    float e1 = expf(v1 - m);
    float s = e0 + e1;
#pragma unroll
    for (int off = 16; off > 0; off >>= 1)
        s += __shfl_xor(s, off, 32);
    float r = 1.0f / s;
    float* o = out + (size_t)node * 64;
    o[lane]      = e0 * r;
    o[lane + 32] = e1 * r;
}

// ---------------------------------------------------------------------------
// Host-side launch. Inputs: x, src, dst, Wl1, Wr1, b1, Wl2, Wr2, b2.
// ---------------------------------------------------------------------------
extern "C" void kernel_launch(void* const* d_in, const int* in_sizes, int n_in,
                              void* d_out, int out_size, void* d_ws, size_t ws_size,
                              hipStream_t stream) {
    const float* x   = (const float*)d_in[0];
    const int*   src = (const int*)  d_in[1];
    const int*   dst = (const int*)  d_in[2];
    const float* Wl1 = (const float*)d_in[3];
    const float* Wr1 = (const float*)d_in[4];
    const float* b1  = (const float*)d_in[5];
    const float* Wl2 = (const float*)d_in[6];
    const float* Wr2 = (const float*)d_in[7];
    const float* b2  = (const float*)d_in[8];
    float* out = (float*)d_out;

    // Workspace layout (floats): [deg N][sum1 N*64][sum2 N*128][h N*128]
    // pre-softmax logits alias sum1 (dead after layer-1 GEMM).
    float* ws   = (float*)d_ws;
    float* deg  = ws;
    float* sum1 = deg  + (size_t)N_NODES;
    float* sum2 = sum1 + (size_t)N_NODES * D_IN;
    float* h    = sum2 + (size_t)N_NODES * D_HID;
    float* pre2 = sum1;

    // 1) zero deg + sum1 + sum2 (contiguous: N*(1+64+128) floats, /4 exact)
    {
        long long n4 = (long long)N_NODES * (1 + D_IN + D_HID) / 4;
        int blocks = (int)((n4 + 255) / 256);
        zero_f4<<<blocks, 256, 0, stream>>>((float4*)ws, n4);
    }
    // 2) degrees (E = 6250 * 256 exactly)
    degree_kernel<<<N_EDGES / 256, 256, 0, stream>>>(dst, deg);
    // 3) layer-1 scatter: E * (64/4) threads = 100000 blocks exactly
    scatter_add_kernel<D_IN><<<(int)((long long)N_EDGES * (D_IN / 4) / 256), 256, 0, stream>>>(
        x, src, dst, sum1);
    // 4) h = relu(agg1 @ Wl1 + x @ Wr1 + b1): 6250*8 tiles / 8 waves = 6250 blocks
    sage_gemm_kernel<D_IN, D_HID, true><<<(N_NODES / 16) * (D_HID / 16) / 8, 256, 0, stream>>>(
        sum1, deg, x, Wl1, Wr1, b1, h);
    // 5) layer-2 scatter: E * (128/4) threads = 200000 blocks exactly
    scatter_add_kernel<D_HID><<<(int)((long long)N_EDGES * (D_HID / 4) / 256), 256, 0, stream>>>(
        h, src, dst, sum2);
    // 6) pre2 = agg2 @ Wl2 + h @ Wr2 + b2: 6250*4 tiles / 8 waves = 3125 blocks
    sage_gemm_kernel<D_HID, D_OUT, false><<<(N_NODES / 16) * (D_OUT / 16) / 8, 256, 0, stream>>>(
        sum2, deg, h, Wl2, Wr2, b2, pre2);
    // 7) softmax rows: 100000 waves / 8 = 12500 blocks
    softmax64_kernel<<<N_NODES / 8, 256, 0, stream>>>(pre2, out);
}